// ObjectAggregation_86354612453536
// MI455X (gfx1250) — compile-verified
//
#include <hip/hip_runtime.h>

#define B_    64
#define OBJS_ 2048
#define N_    (B_ * OBJS_)
#define D_    256
#define K_    2
#define CH_   16          // rows per async-staged chunk in k_pool

typedef __attribute__((ext_vector_type(2))) float v2f;
typedef __attribute__((ext_vector_type(8))) float v8f;

__device__ __forceinline__ float lrelu(float v) { return v >= 0.f ? v : 0.2f * v; }

__device__ __forceinline__ float wave_sum32(float v) {
#pragma unroll
    for (int m = 16; m >= 1; m >>= 1) v += __shfl_xor(v, m, 32);
    return v;
}

// ---------- Kernel 1: per-scene mean ctx[b,d] ----------
__global__ __launch_bounds__(256) void k_ctx(const float* __restrict__ x,
                                             const int* __restrict__ num_objs,
                                             float* __restrict__ ctx) {
    const int b = blockIdx.x;
    const int d = threadIdx.x;
    const float* xs = x + (size_t)b * OBJS_ * D_ + d;
    float acc = 0.f;
    for (int r = 0; r < OBJS_; ++r) {
        if ((r & 63) == 0) __builtin_prefetch(xs + (size_t)(r + 64) * D_, 0, 0);
        acc += xs[(size_t)r * D_];
    }
    int c = num_objs[b];
    float cnt = (float)(c < 1 ? 1 : c);
    ctx[b * D_ + d] = acc / cnt;
}

// ---------- Kernel 2: base[n] = leaky_relu(x+ctx).att_shared, plus per-scene min/max ----------
__global__ __launch_bounds__(256) void k_base(const float* __restrict__ x,
                                              const float* __restrict__ ctx,
                                              const float* __restrict__ att_shared,
                                              float* __restrict__ base,
                                              float* __restrict__ minb,
                                              float* __restrict__ maxb) {
    __shared__ float ctx_s[D_];
    __shared__ float a_s[D_];
    __shared__ float redmin[8], redmax[8];
    const int b = blockIdx.x, t = threadIdx.x;
    ctx_s[t] = ctx[b * D_ + t];
    a_s[t]   = att_shared[t];
    __syncthreads();

    const int lane = t & 31, wid = t >> 5;          // 8 waves of 32
    const int d0 = lane * 8;                        // each lane covers 8 columns
    const float* xs = x + (size_t)b * OBJS_ * D_;
    float mn = 3.4e38f, mx = -3.4e38f;

    for (int i = 0; i < OBJS_ / 8; ++i) {
        const int r = wid * (OBJS_ / 8) + i;
        const float4* xr = (const float4*)(xs + (size_t)r * D_ + d0);
        float4 xa = xr[0], xb = xr[1];
        float xv[8] = {xa.x, xa.y, xa.z, xa.w, xb.x, xb.y, xb.z, xb.w};
        float p = 0.f;
#pragma unroll
        for (int j = 0; j < 8; ++j)
            p += lrelu(xv[j] + ctx_s[d0 + j]) * a_s[d0 + j];
        float s = wave_sum32(p);
        if (lane == 0) base[b * OBJS_ + r] = s;
        mn = fminf(mn, s);
        mx = fmaxf(mx, s);
    }
    if (lane == 0) { redmin[wid] = mn; redmax[wid] = mx; }
    __syncthreads();
    if (t == 0) {
        float m1 = redmin[0], m2 = redmax[0];
#pragma unroll
        for (int i = 1; i < 8; ++i) { m1 = fminf(m1, redmin[i]); m2 = fmaxf(m2, redmax[i]); }
        minb[b] = m1; maxb[b] = m2;
    }
}

// ---------- Kernel 3: per-scene softmax stats m[b,k], s[b,k]; bias_term ----------
__global__ __launch_bounds__(256) void k_stats(const float* __restrict__ base,
                                               const float* __restrict__ att_shared,
                                               const float* __restrict__ att_scale,
                                               const float* __restrict__ channel_bias,
                                               const float* __restrict__ minb,
                                               const float* __restrict__ maxb,
                                               float* __restrict__ mbuf,
                                               float* __restrict__ sbuf,
                                               float* __restrict__ btbuf) {
    __shared__ float red0[256], red1[256];
    __shared__ float bc[6];
    const int b = blockIdx.x, t = threadIdx.x;

    // bias_term[k] = channel_bias[k,:].att_shared
    float a = att_shared[t];
    red0[t] = channel_bias[t] * a;
    red1[t] = channel_bias[D_ + t] * a;
    __syncthreads();
#pragma unroll
    for (int off = 128; off >= 1; off >>= 1) {
        if (t < off) { red0[t] += red0[t + off]; red1[t] += red1[t + off]; }
        __syncthreads();
    }
    if (t == 0) {
        float bt0 = red0[0], bt1 = red1[0];
        float sc0 = att_scale[0], sc1 = att_scale[1];
        float mnb = minb[b], mxb = maxb[b];
        float m0 = (sc0 >= 0.f ? (mxb + bt0) : (mnb + bt0)) * sc0;
        float m1 = (sc1 >= 0.f ? (mxb + bt1) : (mnb + bt1)) * sc1;
        bc[0] = bt0; bc[1] = bt1; bc[2] = m0; bc[3] = m1; bc[4] = sc0; bc[5] = sc1;
    }
    __syncthreads();
    const float bt0 = bc[0], bt1 = bc[1], m0 = bc[2], m1 = bc[3], sc0 = bc[4], sc1 = bc[5];

    float s0 = 0.f, s1 = 0.f;
    const float* bp = base + b * OBJS_;
    for (int i = 0; i < OBJS_ / 256; ++i) {
        float bv = bp[t + i * 256];
        s0 += expf((bv + bt0) * sc0 - m0);
        s1 += expf((bv + bt1) * sc1 - m1);
    }
    red0[t] = s0; red1[t] = s1;
    __syncthreads();
#pragma unroll
    for (int off = 128; off >= 1; off >>= 1) {
        if (t < off) { red0[t] += red0[t + off]; red1[t] += red1[t + off]; }
        __syncthreads();
    }
    if (t == 0) {
        mbuf[b * K_ + 0] = m0; mbuf[b * K_ + 1] = m1;
        sbuf[b * K_ + 0] = red0[0]; sbuf[b * K_ + 1] = red1[0];
        if (b == 0) { btbuf[0] = bt0; btbuf[1] = bt1; }
    }
}

// ---------- Kernel 4: weights + WMMA pooling  S[b] = X^T (256x2048) * W (2048x2) ----------
// x tile is staged chunk-by-chunk into LDS with async global->LDS copies (double buffered).
__global__ __launch_bounds__(512) void k_pool(const float* __restrict__ x,
                                              const float* __restrict__ base,
                                              const float* __restrict__ mbuf,
                                              const float* __restrict__ sbuf,
                                              const float* __restrict__ btbuf,
                                              const float* __restrict__ att_scale,
                                              float* __restrict__ out_feat,
                                              float* __restrict__ out_w) {
    __shared__ float wch[K_ * OBJS_];                    // channel-major weights, 16 KB
    __shared__ __align__(16) float xbuf[2][CH_ * D_];    // 2 x 16 KB staging buffers
    const int b = blockIdx.x, t = threadIdx.x;

    // ---- Phase A: attention weights -> LDS (channel-major) and global output ----
    const float bt0 = btbuf[0], bt1 = btbuf[1];
    const float m0 = mbuf[b * K_ + 0], m1 = mbuf[b * K_ + 1];
    const float is0 = 1.f / sbuf[b * K_ + 0], is1 = 1.f / sbuf[b * K_ + 1];
    const float sc0 = att_scale[0], sc1 = att_scale[1];
    const float* bp = base + b * OBJS_;
    float* wp = out_w + (size_t)b * OBJS_ * K_;
#pragma unroll
    for (int i = 0; i < OBJS_ / 512; ++i) {
        const int r = t + i * 512;
        float bv = bp[r];
        float w0 = expf((bv + bt0) * sc0 - m0) * is0;
        float w1 = expf((bv + bt1) * sc1 - m1) * is1;
        wch[r]         = w0;
        wch[OBJS_ + r] = w1;
        *(float2*)&wp[r * K_] = make_float2(w0, w1);
    }

    // ---- Async staging of x chunks ----
    const float* xs = x + (size_t)b * OBJS_ * D_;
    auto stage = [&](int c, int buf) {
#pragma unroll
        for (int j = 0; j < (CH_ * D_ / 4) / 512; ++j) {       // 2 x b128 per thread
            const int idx = t + 512 * j;                       // float4 index
            unsigned lds_off = (unsigned)(uintptr_t)(&xbuf[buf][idx * 4]);
            unsigned long long ga =
                (unsigned long long)(uintptr_t)(xs + (size_t)c * CH_ * D_ + idx * 4);
            asm volatile("global_load_async_to_lds_b128 %0, %1, off"
                         :: "v"(lds_off), "v"(ga) : "memory");
        }
    };

    __syncthreads();                 // wch visible before compute; xbuf free
    stage(0, 0);
    asm volatile("s_wait_asynccnt 0x0" ::: "memory");
    __syncthreads();

    // ---- Phase B: WMMA f32 16x16x4. Wave 'wid' owns d-tile [16*wid, 16*wid+16). ----
    const int lane = t & 31, wid = t >> 5;       // 16 waves
    const int d_base = wid * 16;
    const int nc  = lane & 15;                   // A: M index (d); B: N index (channel col)
    const int hi  = lane >> 4;                   // half-wave: K pair select
    const bool act = nc < K_;
    const int ncc = act ? nc : 0;                // clamped channel for unconditional LDS load
    v8f acc0 = {0.f, 0.f, 0.f, 0.f, 0.f, 0.f, 0.f, 0.f};
    v8f acc1 = {0.f, 0.f, 0.f, 0.f, 0.f, 0.f, 0.f, 0.f};

    const int NCHUNK = OBJS_ / CH_;              // 128
    for (int c = 0; c < NCHUNK; ++c) {
        if (c + 1 < NCHUNK) stage(c + 1, (c + 1) & 1);

        const float* xb = xbuf[c & 1];
        const int rg = c * CH_;                  // global row base of this chunk
#pragma unroll
        for (int s = 0; s < CH_ / 4; ++s) {
            const int rl = s * 4 + 2 * hi;       // local row (A/B K-pair base)
            v2f av, bv;
            av.x = xb[rl * D_ + d_base + nc];            // A(m=nc, k=2*hi)
            av.y = xb[(rl + 1) * D_ + d_base + nc];      // A(m=nc, k=2*hi+1)
            const float2 wv = *(const float2*)&wch[ncc * OBJS_ + rg + rl];  // b64, aligned
            bv.x = act ? wv.x : 0.f;                     // cndmask, not exec-predication
            bv.y = act ? wv.y : 0.f;
            if (s & 1)
                acc1 = __builtin_amdgcn_wmma_f32_16x16x4_f32(false, av, false, bv,
                                                             (short)0, acc1, false, false);
            else
                acc0 = __builtin_amdgcn_wmma_f32_16x16x4_f32(false, av, false, bv,
                                                             (short)0, acc0, false, false);
        }
        asm volatile("s_wait_asynccnt 0x0" ::: "memory");
        __syncthreads();
    }

    if (act) {
        // C/D layout: lane N=nc, VGPR r -> M = r + 8*hi
#pragma unroll
        for (int r = 0; r < 8; ++r) {
            out_feat[(size_t)b * D_ * K_ + (size_t)(d_base + 8 * hi + r) * K_ + nc] =
                acc0[r] + acc1[r];
        }
    }
}

extern "C" void kernel_launch(void* const* d_in, const int* in_sizes, int n_in,
                              void* d_out, int out_size, void* d_ws, size_t ws_size,
                              hipStream_t stream) {
    const float* x            = (const float*)d_in[0];
    const int*   num_objs     = (const int*)d_in[1];
    const float* att_shared   = (const float*)d_in[2];
    const float* att_scale    = (const float*)d_in[3];
    const float* channel_bias = (const float*)d_in[4];

    float* out_feat = (float*)d_out;                 // [B, D, K]
    float* out_w    = out_feat + B_ * D_ * K_;       // [N, K]

    float* ws    = (float*)d_ws;
    float* ctx   = ws;                   // B*D
    float* base  = ctx + B_ * D_;        // N
    float* minb  = base + N_;            // B
    float* maxb  = minb + B_;            // B
    float* mbuf  = maxb + B_;            // B*K
    float* sbuf  = mbuf + B_ * K_;       // B*K
    float* btbuf = sbuf + B_ * K_;       // K

    k_ctx  <<<B_, 256, 0, stream>>>(x, num_objs, ctx);
    k_base <<<B_, 256, 0, stream>>>(x, ctx, att_shared, base, minb, maxb);
    k_stats<<<B_, 256, 0, stream>>>(base, att_shared, att_scale, channel_bias,
                                    minb, maxb, mbuf, sbuf, btbuf);
    k_pool <<<B_, 512, 0, stream>>>(x, base, mbuf, sbuf, btbuf, att_scale,
                                    out_feat, out_w);
}